// AttentionFusion_64553358458977
// MI455X (gfx1250) — compile-verified
//
#include <hip/hip_runtime.h>

// Problem constants (from reference)
#define PD   1024
#define MD   512
#define HH   512
#define BB   256
#define VV   2048
#define W1LD (PD + MD)   // W1 row length = 1536

typedef __attribute__((ext_vector_type(2))) float v2f;
typedef __attribute__((ext_vector_type(8))) float v8f;

// Single-instruction tanh: gfx1250 has V_TANH_F32 (TRANS op). Prefer the
// clang builtin; otherwise emit the instruction directly so we never fall
// back to the OCML polynomial in the 268M-tanh hot loop.
__device__ __forceinline__ float fast_tanh(float x) {
#if __has_builtin(__builtin_amdgcn_tanhf)
    return __builtin_amdgcn_tanhf(x);
#else
    float r;
    asm("v_tanh_f32 %0, %1" : "=v"(r) : "v"(x));
    return r;
#endif
}

// ---------------------------------------------------------------------------
// WMMA f32 GEMM: out[m,n] = sum_k A[m*LDA + k] * W1[n*W1LD + KOFF + k] (+bias[n])
// One wave computes one 16x16 tile using V_WMMA_F32_16X16X4_F32.
// f32 A-matrix 16x4 layout: lane l holds M=l&15; VGPR0 = K(2*(l>>4)), VGPR1 = K+1.
// f32 B-matrix 4x16 layout mirrors it: lane l holds N=l&15; VGPR0/1 = rows K,K+1.
// C/D: VGPR r, lane l -> D[r + 8*(l>>4)][l&15].
// ---------------------------------------------------------------------------
template<int KDIM, int LDA, int KOFF, bool BIAS>
__global__ __launch_bounds__(256) void wmma_gemm_w1t(
        const float* __restrict__ A,
        const float* __restrict__ W1,
        const float* __restrict__ bias,
        float* __restrict__ out) {
    const int wid  = blockIdx.x * 8 + (threadIdx.x >> 5); // 8 waves / block
    const int lane = threadIdx.x & 31;
    const int half = lane >> 4;     // 0 or 1
    const int lm   = lane & 15;
    const int tm   = wid >> 5;      // tiles along N = 512/16 = 32
    const int tn   = wid & 31;
    const int m0   = tm * 16, n0 = tn * 16;

    const float* arow = A  + (size_t)(m0 + lm) * LDA  + 2 * half;
    const float* brow = W1 + (size_t)(n0 + lm) * W1LD + KOFF + 2 * half;

    v8f acc = {};
#pragma unroll 4
    for (int k = 0; k < KDIM; k += 4) {
        v2f a = *(const v2f*)(arow + k);   // A[m][k+2h], A[m][k+2h+1]
        v2f b = *(const v2f*)(brow + k);   // Bt: W1[n][.. k+2h], k+2h+1
        acc = __builtin_amdgcn_wmma_f32_16x16x4_f32(
                  false, a, false, b, (short)0, acc, false, false);
    }

    const float bv = BIAS ? bias[n0 + lm] : 0.0f;
#pragma unroll
    for (int r = 0; r < 8; ++r) {
        const int row = m0 + r + 8 * half;
        out[(size_t)row * HH + n0 + lm] = acc[r] + bv;
    }
}

// ---------------------------------------------------------------------------
// Fused kernel: scores[b,v] = b2 + sum_h tanh(hp[b,h] + hm[v,h]) * w2[h]
// 64x64 (b x v) tile per block, H chunked by 64 through LDS.
// Thread micro-tile 4x4 with rows strided by 16 -> conflict-free b128 LDS reads
// (row stride 68 floats: (tv + 16j)*68 mod 64 == 4*tv -> all 16 banks distinct).
// ---------------------------------------------------------------------------
#define TB  64
#define TV  64
#define HC  64
#define LDT 68   // 64 + 4 pad, keeps 16B alignment for float4

__global__ __launch_bounds__(256) void fused_tanh_score(
        const float* __restrict__ hp, const float* __restrict__ hm,
        const float* __restrict__ w2, const float* __restrict__ b2,
        float* __restrict__ out) {
    __shared__ float hps[TB * LDT];
    __shared__ float hms[TV * LDT];
    __shared__ float w2s[HC];

    const int tid = threadIdx.x;
    const int b0  = blockIdx.y * TB;
    const int v0  = blockIdx.x * TV;
    const int tb  = tid >> 4;   // 0..15
    const int tv  = tid & 15;   // 0..15

    float acc[4][4] = {};

    for (int c = 0; c < HH; c += HC) {
        __syncthreads();  // WAR: previous chunk's readers done
        // Stage hp/hm 64x64 chunks (coalesced float4 global -> float4 LDS)
#pragma unroll
        for (int i = 0; i < 4; ++i) {          // 1024 float4 / 256 threads
            const int idx = tid + i * 256;
            const int row = idx >> 4;          // 16 float4 per row
            const int c4  = (idx & 15) * 4;
            float4 p = *(const float4*)(hp + (size_t)(b0 + row) * HH + c + c4);
            float4 q = *(const float4*)(hm + (size_t)(v0 + row) * HH + c + c4);
            *(float4*)&hps[row * LDT + c4] = p;
            *(float4*)&hms[row * LDT + c4] = q;
        }
        if (tid < HC) w2s[tid] = w2[c + tid];
        __syncthreads();

#pragma unroll 4
        for (int h = 0; h < HC; h += 4) {
            const float4 wq = *(const float4*)&w2s[h];
            float4 pq[4], mq[4];
#pragma unroll
            for (int i = 0; i < 4; ++i)
                pq[i] = *(const float4*)&hps[(tb + 16 * i) * LDT + h];
#pragma unroll
            for (int j = 0; j < 4; ++j)
                mq[j] = *(const float4*)&hms[(tv + 16 * j) * LDT + h];
#pragma unroll
            for (int i = 0; i < 4; ++i)
#pragma unroll
                for (int j = 0; j < 4; ++j) {
                    acc[i][j] += fast_tanh(pq[i].x + mq[j].x) * wq.x;
                    acc[i][j] += fast_tanh(pq[i].y + mq[j].y) * wq.y;
                    acc[i][j] += fast_tanh(pq[i].z + mq[j].z) * wq.z;
                    acc[i][j] += fast_tanh(pq[i].w + mq[j].w) * wq.w;
                }
        }
    }

    const float bias2 = b2[0];
#pragma unroll
    for (int i = 0; i < 4; ++i)
#pragma unroll
        for (int j = 0; j < 4; ++j)
            out[(size_t)(b0 + tb + 16 * i) * VV + (v0 + tv + 16 * j)]
                = acc[i][j] + bias2;
}

// ---------------------------------------------------------------------------
extern "C" void kernel_launch(void* const* d_in, const int* in_sizes, int n_in,
                              void* d_out, int out_size, void* d_ws, size_t ws_size,
                              hipStream_t stream) {
    const float* patient = (const float*)d_in[0];  // (256,1024)
    const float* atc4    = (const float*)d_in[1];  // (2048,512)
    const float* W1      = (const float*)d_in[2];  // (512,1536)
    const float* b1      = (const float*)d_in[3];  // (512,)
    const float* w2      = (const float*)d_in[4];  // (512,)
    const float* b2      = (const float*)d_in[5];  // (1,)
    float* scores = (float*)d_out;                 // (256,2048)

    float* hp = (float*)d_ws;                      // 256*512 f32 (+b1 folded in)
    float* hm = hp + (size_t)BB * HH;              // 2048*512 f32

    // hp = patient @ W1[:, :PD]^T + b1 : 512 tiles -> 64 blocks x 8 waves
    hipLaunchKernelGGL((wmma_gemm_w1t<PD, PD, 0, true>),
                       dim3(64), dim3(256), 0, stream, patient, W1, b1, hp);
    // hm = atc4 @ W1[:, PD:]^T        : 4096 tiles -> 512 blocks x 8 waves
    hipLaunchKernelGGL((wmma_gemm_w1t<MD, MD, PD, false>),
                       dim3(512), dim3(256), 0, stream, atc4, W1, nullptr, hm);
    // fused tanh + reduce over h
    hipLaunchKernelGGL(fused_tanh_score,
                       dim3(VV / TV, BB / TB), dim3(256), 0, stream,
                       hp, hm, w2, b2, scores);
}